// BEVFormerEncoder_2860448219634
// MI455X (gfx1250) — compile-verified
//
#include <hip/hip_runtime.h>

#define Q_TOT 22500
#define NCAM  6
#define DPTH  4
#define CCH   128
#define FH    48
#define FW    88

typedef __attribute__((ext_vector_type(2))) float v2f;
typedef __attribute__((ext_vector_type(4))) float v4f;
typedef __attribute__((ext_vector_type(8))) float v8f;

// Block = 128 threads (4 wave32), handles 16 queries.
// Wave 0: WMMA projection  D[(cam,row), q] = l2i(18x4) x rp4(4x16)  via 2x V_WMMA_F32_16X16X4_F32 per depth.
// All waves: gather phase, thread t -> query (t&15), channel block (t>>4)*16.
__global__ __launch_bounds__(128) void bev_sample_kernel(
    const float* __restrict__ feat,   // (NCAM, FH, FW, C)
    const float* __restrict__ Imat,   // (NCAM, 3, 3)
    const float* __restrict__ Emat,   // (NCAM, 4, 4)
    const float* __restrict__ grid,   // (D, 3, Q)
    float* __restrict__ out)          // (Q, C)
{
    // [d][q][combo]  combo = cam*3 + row (0..17 valid, padded to 32 for D1 spill)
    __shared__ float projLDS[DPTH][16][32];

    const int tid = threadIdx.x;
    const int q0  = blockIdx.x * 16;

    if (tid < 32) {
        const int lane = tid;
        const int m    = lane & 15;      // A-matrix row within tile
        const int hi   = lane >> 4;      // 0: K={0,1}, 1: K={2,3}
        const int j0   = 2 * hi;         // first K column this lane holds

        // A0: rows = combos 0..15 ; A1: rows = combos 16..17 (rest zero)
        v2f a0, a1;
        {
            int n = m / 3, i = m - 3 * n;
            float s0 = 0.f, s1 = 0.f;
            #pragma unroll
            for (int k = 0; k < 3; ++k) {
                float Iv = Imat[n * 9 + i * 3 + k];
                s0 += Iv * Emat[n * 16 + k * 4 + j0];
                s1 += Iv * Emat[n * 16 + k * 4 + j0 + 1];
            }
            a0.x = s0; a0.y = s1;
        }
        {
            int combo = 16 + m;
            float s0 = 0.f, s1 = 0.f;
            if (combo < 18) {
                int n = 5, i = combo - 15;
                #pragma unroll
                for (int k = 0; k < 3; ++k) {
                    float Iv = Imat[n * 9 + i * 3 + k];
                    s0 += Iv * Emat[n * 16 + k * 4 + j0];
                    s1 += Iv * Emat[n * 16 + k * 4 + j0 + 1];
                }
            }
            a1.x = s0; a1.y = s1;
        }

        int gq = q0 + m;
        if (gq > Q_TOT - 1) gq = Q_TOT - 1;

        for (int d = 0; d < DPTH; ++d) {
            // B[k, q] = rp4 component k of query q (lane holds K = {j0, j0+1})
            float gx = grid[(d * 3 + 0) * Q_TOT + gq] * 102.4f - 51.2f;
            float gy = grid[(d * 3 + 1) * Q_TOT + gq] * 102.4f - 51.2f;
            float gz = grid[(d * 3 + 2) * Q_TOT + gq] * 8.0f   - 5.0f;
            v2f b;
            b.x = hi ? gz   : gx;
            b.y = hi ? 1.0f : gy;

            v8f c0 = {}; v8f c1 = {};
            v8f d0 = __builtin_amdgcn_wmma_f32_16x16x4_f32(
                false, a0, false, b, (short)0, c0, false, false);
            v8f d1 = __builtin_amdgcn_wmma_f32_16x16x4_f32(
                false, a1, false, b, (short)0, c1, false, false);

            // D layout: VGPR r, lane l -> M = r + 8*(l>=16), N = l&15
            float* dst0 = &projLDS[d][m][8 * hi];        // combos 0..15
            float* dst1 = &projLDS[d][m][16 + 8 * hi];   // combos 16..23 / pad 24..31
            #pragma unroll
            for (int r = 0; r < 8; ++r) dst0[r] = d0[r];
            #pragma unroll
            for (int r = 0; r < 8; ++r) dst1[r] = d1[r];
        }
    }
    __syncthreads();

    // ---------------- gather / accumulate phase ----------------
    const int qs = tid & 15;          // query slot
    const int cb = tid >> 4;          // channel block (16 channels)
    const int q  = q0 + qs;

    float acc[16];
    #pragma unroll
    for (int j = 0; j < 16; ++j) acc[j] = 0.f;
    float cnt = 0.f;

    for (int n = 0; n < NCAM; ++n) {
        const float* fbase = feat + (size_t)n * FH * FW * CCH + cb * 16;
        #pragma unroll
        for (int d = 0; d < DPTH; ++d) {
            float xw = projLDS[d][qs][n * 3 + 0];
            float yw = projLDS[d][qs][n * 3 + 1];
            float zc = projLDS[d][qs][n * 3 + 2];
            float zs = fmaxf(zc, 1e-5f);
            float u  = xw / (zs * 800.0f);
            float v  = yw / (zs * 480.0f);
            bool mask = (zc > 1e-5f) & (u > 0.f) & (u < 1.f) & (v > 0.f) & (v < 1.f);
            float mf = mask ? 1.f : 0.f;
            cnt += mf;

            float px  = u * (float)FW - 0.5f;
            float py  = v * (float)FH - 0.5f;
            float x0f = floorf(px), y0f = floorf(py);
            float wx1 = px - x0f,   wy1 = py - y0f;
            float wx0 = 1.f - wx1,  wy0 = 1.f - wy1;
            int x0i = (int)x0f, y0i = (int)y0f;

            #pragma unroll
            for (int cy = 0; cy < 2; ++cy) {
                int   yi = y0i + cy;
                float wy = cy ? wy1 : wy0;
                bool  vy = (yi >= 0) & (yi < FH);
                int   yc = yi < 0 ? 0 : (yi > FH - 1 ? FH - 1 : yi);
                #pragma unroll
                for (int cx = 0; cx < 2; ++cx) {
                    int   xi = x0i + cx;
                    float wx = cx ? wx1 : wx0;
                    bool  vx = (xi >= 0) & (xi < FW);
                    int   xc = xi < 0 ? 0 : (xi > FW - 1 ? FW - 1 : xi);
                    float w  = mf * wy * wx * ((vy & vx) ? 1.f : 0.f);
                    const v4f* p = (const v4f*)(fbase + (size_t)(yc * FW + xc) * CCH);
                    #pragma unroll
                    for (int j = 0; j < 4; ++j) {
                        v4f f4 = p[j];
                        acc[4 * j + 0] += w * f4.x;
                        acc[4 * j + 1] += w * f4.y;
                        acc[4 * j + 2] += w * f4.z;
                        acc[4 * j + 3] += w * f4.w;
                    }
                }
            }
        }
    }

    float inv = 1.f / fmaxf(cnt, 1.f);
    if (q < Q_TOT) {
        v4f* op = (v4f*)(out + (size_t)q * CCH + cb * 16);
        #pragma unroll
        for (int j = 0; j < 4; ++j) {
            v4f o;
            o.x = acc[4 * j + 0] * inv;
            o.y = acc[4 * j + 1] * inv;
            o.z = acc[4 * j + 2] * inv;
            o.w = acc[4 * j + 3] * inv;
            op[j] = o;
        }
    }
}

extern "C" void kernel_launch(void* const* d_in, const int* in_sizes, int n_in,
                              void* d_out, int out_size, void* d_ws, size_t ws_size,
                              hipStream_t stream) {
    (void)in_sizes; (void)n_in; (void)out_size; (void)d_ws; (void)ws_size;
    const float* feat = (const float*)d_in[0];  // (1,6,48,88,128)
    const float* I    = (const float*)d_in[1];  // (1,6,3,3)
    const float* E    = (const float*)d_in[2];  // (1,6,4,4)
    const float* grid = (const float*)d_in[3];  // (4,3,150,150)
    float* out = (float*)d_out;                 // (1,22500,128)

    const int nblocks = (Q_TOT + 15) / 16;      // 1407
    bev_sample_kernel<<<nblocks, 128, 0, stream>>>(feat, I, E, grid, out);
}